// SVD_Weighted_6262062318062
// MI455X (gfx1250) — compile-verified
//
#include <hip/hip_runtime.h>
#include <cstdint>

typedef float v2f __attribute__((ext_vector_type(2)));
typedef float v8f __attribute__((ext_vector_type(8)));

#define BATCHES 8192
#define NPTS    2048
#define NT      256
#define NV4     (NPTS / 4)   // 512 float4 per row

__global__ __launch_bounds__(NT) void kabsch_wsvd_kernel(
    const float* __restrict__ src,   // (B,3,N)
    const float* __restrict__ corr,  // (B,3,N)
    const float* __restrict__ wts,   // (B,N)
    float* __restrict__ rot_out,     // (B,3,3) flat
    float* __restrict__ tr_out)      // (B,3) flat
{
    __shared__ float4 tile[7][NV4];     // rows: w, a0,a1,a2, b0,b1,b2  (56 KiB)
    __shared__ float  partial[8][16];   // per-wave partial sums
    __shared__ float  sums[16];

    const int b   = blockIdx.x;
    const int tid = threadIdx.x;

    const float* baseA = src  + (size_t)b * 3 * NPTS;
    const float* baseB = corr + (size_t)b * 3 * NPTS;
    const float* rows[7] = { wts + (size_t)b * NPTS,
                             baseA, baseA + NPTS, baseA + 2 * NPTS,
                             baseB, baseB + NPTS, baseB + 2 * NPTS };

    // ---- Stage the whole batch tile into LDS via CDNA5 async-tensor path ----
    #pragma unroll
    for (int r = 0; r < 7; ++r) {
        const float4* gp = (const float4*)rows[r];
        {
            uint32_t la = (uint32_t)(uintptr_t)&tile[r][tid];
            uint64_t ga = (uint64_t)(uintptr_t)(gp + tid);
            asm volatile("global_load_async_to_lds_b128 %0, %1, off"
                         :: "v"(la), "v"(ga) : "memory");
        }
        {
            uint32_t la = (uint32_t)(uintptr_t)&tile[r][tid + NT];
            uint64_t ga = (uint64_t)(uintptr_t)(gp + tid + NT);
            asm volatile("global_load_async_to_lds_b128 %0, %1, off"
                         :: "v"(la), "v"(ga) : "memory");
        }
    }
    asm volatile("s_wait_asynccnt 0" ::: "memory");
    __syncthreads();

    // ---- One-pass sufficient statistics: 16 accumulators per lane ----
    // acc[0]=Sw, acc[1..3]=Sum w*a_d, acc[4..6]=Sum w*b_d, acc[7+3d+e]=Sum w*a_d*b_e
    float acc[16];
    #pragma unroll
    for (int k = 0; k < 16; ++k) acc[k] = 0.0f;

    #pragma unroll
    for (int ch = 0; ch < 2; ++ch) {
        const int c = tid + ch * NT;
        float4 w4 = tile[0][c];
        float4 xa = tile[1][c], ya = tile[2][c], za = tile[3][c];
        float4 xb = tile[4][c], yb = tile[5][c], zb = tile[6][c];
        const float* wp = (const float*)&w4;
        const float* A0 = (const float*)&xa;
        const float* A1 = (const float*)&ya;
        const float* A2 = (const float*)&za;
        const float* B0 = (const float*)&xb;
        const float* B1 = (const float*)&yb;
        const float* B2 = (const float*)&zb;
        #pragma unroll
        for (int j = 0; j < 4; ++j) {
            float wv = wp[j];
            float a0 = A0[j], a1 = A1[j], a2 = A2[j];
            float b0 = B0[j], b1 = B1[j], b2 = B2[j];
            float wa0 = wv * a0, wa1 = wv * a1, wa2 = wv * a2;
            acc[0]  += wv;
            acc[1]  += wa0;      acc[2]  += wa1;      acc[3]  += wa2;
            acc[4]  += wv * b0;  acc[5]  += wv * b1;  acc[6]  += wv * b2;
            acc[7]  += wa0 * b0; acc[8]  += wa0 * b1; acc[9]  += wa0 * b2;
            acc[10] += wa1 * b0; acc[11] += wa1 * b1; acc[12] += wa1 * b2;
            acc[13] += wa2 * b0; acc[14] += wa2 * b1; acc[15] += wa2 * b2;
        }
    }

    // ---- Intra-wave butterfly reduction (wave32) ----
    #pragma unroll
    for (int off = 16; off > 0; off >>= 1) {
        #pragma unroll
        for (int k = 0; k < 16; ++k) acc[k] += __shfl_xor(acc[k], off, 32);
    }
    const int wave = tid >> 5, lane = tid & 31;
    if (lane == 0) {
        #pragma unroll
        for (int k = 0; k < 16; ++k) partial[wave][k] = acc[k];
    }
    __syncthreads();

    // ---- Cross-wave reduction on the matrix engine: D = P(16x4) * ones_col ----
    if (tid < 32) {
        const int m  = tid & 15;
        const int kb = (tid < 16) ? 0 : 2;   // A-matrix f32 layout: lanes 0-15 hold K=0,1; 16-31 hold K=2,3
        v2f a1; a1[0] = partial[kb + 0][m]; a1[1] = partial[kb + 1][m];   // waves 0..3
        v2f a2; a2[0] = partial[kb + 4][m]; a2[1] = partial[kb + 5][m];   // waves 4..7
        v2f bo; { float one = (m == 0) ? 1.0f : 0.0f; bo[0] = one; bo[1] = one; }
        v8f cc = {};
        cc = __builtin_amdgcn_wmma_f32_16x16x4_f32(false, a1, false, bo, (short)0, cc, false, false);
        cc = __builtin_amdgcn_wmma_f32_16x16x4_f32(false, a2, false, bo, (short)0, cc, false, false);
        // D column N=0: lane 0 holds M=0..7 in c[0..7], lane 16 holds M=8..15
        if (tid == 0) {
            #pragma unroll
            for (int i = 0; i < 8; ++i) sums[i] = cc[i];
        }
        if (tid == 16) {
            #pragma unroll
            for (int i = 0; i < 8; ++i) sums[8 + i] = cc[i];
        }
    }

    // ---- Per-batch 3x3 solve on lane 0 (same wave as the WMMA: no barrier needed) ----
    if (tid == 0) {
        const float Sw  = sums[0];
        const float Dnm = Sw + 1e-5f;
        const float inv = 1.0f / Dnm;
        float ca[3], cb[3];
        #pragma unroll
        for (int d = 0; d < 3; ++d) { ca[d] = sums[1 + d] * inv; cb[d] = sums[4 + d] * inv; }
        const float two_t = 2.0f - Sw * inv;
        float M[3][3];
        #pragma unroll
        for (int d = 0; d < 3; ++d) {
            #pragma unroll
            for (int e = 0; e < 3; ++e)
                M[d][e] = sums[7 + d * 3 + e] * inv - two_t * ca[d] * cb[e];
        }

        // S = M^T M (symmetric)
        float Smat[3][3];
        #pragma unroll
        for (int i = 0; i < 3; ++i) {
            #pragma unroll
            for (int j = 0; j < 3; ++j)
                Smat[i][j] = M[0][i] * M[0][j] + M[1][i] * M[1][j] + M[2][i] * M[2][j];
        }

        float V[3][3] = { {1.f, 0.f, 0.f}, {0.f, 1.f, 0.f}, {0.f, 0.f, 1.f} };

        auto jrot = [&](int p, int q, int r) {
            float apq = Smat[p][q];
            if (fabsf(apq) < 1e-20f) return;
            float tau = (Smat[q][q] - Smat[p][p]) / (2.0f * apq);
            float tt  = copysignf(1.0f, tau) / (fabsf(tau) + sqrtf(1.0f + tau * tau));
            float c   = rsqrtf(1.0f + tt * tt);
            float sn  = tt * c;
            float spp = Smat[p][p], sqq = Smat[q][q];
            Smat[p][p] = spp - tt * apq;
            Smat[q][q] = sqq + tt * apq;
            Smat[p][q] = Smat[q][p] = 0.0f;
            float srp = Smat[r][p], srq = Smat[r][q];
            Smat[r][p] = Smat[p][r] = c * srp - sn * srq;
            Smat[r][q] = Smat[q][r] = sn * srp + c * srq;
            #pragma unroll
            for (int i = 0; i < 3; ++i) {
                float vp = V[i][p], vq = V[i][q];
                V[i][p] = c * vp - sn * vq;
                V[i][q] = sn * vp + c * vq;
            }
        };
        for (int sweep = 0; sweep < 8; ++sweep) { jrot(0, 1, 2); jrot(0, 2, 1); jrot(1, 2, 0); }

        float lam[3] = { Smat[0][0], Smat[1][1], Smat[2][2] };
        auto csw = [&](int i, int j) {
            if (lam[i] < lam[j]) {
                float t = lam[i]; lam[i] = lam[j]; lam[j] = t;
                #pragma unroll
                for (int r = 0; r < 3; ++r) { float v = V[r][i]; V[r][i] = V[r][j]; V[r][j] = v; }
            }
        };
        csw(0, 1); csw(0, 2); csw(1, 2);   // descending, smallest in column 2

        float U[3][3];
        #pragma unroll
        for (int i = 0; i < 3; ++i) {
            float x = M[0][0] * V[0][i] + M[0][1] * V[1][i] + M[0][2] * V[2][i];
            float y = M[1][0] * V[0][i] + M[1][1] * V[1][i] + M[1][2] * V[2][i];
            float z = M[2][0] * V[0][i] + M[2][1] * V[1][i] + M[2][2] * V[2][i];
            float nn  = x * x + y * y + z * z;
            float ninv = rsqrtf(fmaxf(nn, 1e-30f));
            U[0][i] = x * ninv; U[1][i] = y * ninv; U[2][i] = z * ninv;
        }

        float R[3][3];
        #pragma unroll
        for (int d = 0; d < 3; ++d) {
            #pragma unroll
            for (int e = 0; e < 3; ++e)
                R[d][e] = V[d][0] * U[e][0] + V[d][1] * U[e][1] + V[d][2] * U[e][2];
        }

        float det = R[0][0] * (R[1][1] * R[2][2] - R[1][2] * R[2][1])
                  - R[0][1] * (R[1][0] * R[2][2] - R[1][2] * R[2][0])
                  + R[0][2] * (R[1][0] * R[2][1] - R[1][1] * R[2][0]);
        if (!(det > 0.0f)) {
            #pragma unroll
            for (int d = 0; d < 3; ++d) {
                #pragma unroll
                for (int e = 0; e < 3; ++e)
                    R[d][e] -= 2.0f * V[d][2] * U[e][2];
            }
        }

        float* ro = rot_out + (size_t)b * 9;
        #pragma unroll
        for (int d = 0; d < 3; ++d) {
            #pragma unroll
            for (int e = 0; e < 3; ++e)
                ro[d * 3 + e] = R[d][e];
        }
        float* to = tr_out + (size_t)b * 3;
        #pragma unroll
        for (int d = 0; d < 3; ++d)
            to[d] = cb[d] - (R[d][0] * ca[0] + R[d][1] * ca[1] + R[d][2] * ca[2]);
    }
}

extern "C" void kernel_launch(void* const* d_in, const int* in_sizes, int n_in,
                              void* d_out, int out_size, void* d_ws, size_t ws_size,
                              hipStream_t stream) {
    (void)in_sizes; (void)n_in; (void)out_size; (void)d_ws; (void)ws_size;
    const float* src  = (const float*)d_in[0];
    const float* corr = (const float*)d_in[1];
    const float* wts  = (const float*)d_in[2];
    float* out = (float*)d_out;
    float* rot = out;                           // (B,3,3) flat
    float* tr  = out + (size_t)BATCHES * 9;     // (B,3) flat
    kabsch_wsvd_kernel<<<dim3(BATCHES), dim3(NT), 0, stream>>>(src, corr, wts, rot, tr);
}